// FsodRPN_16277926052052
// MI455X (gfx1250) — compile-verified
//
#include <hip/hip_runtime.h>
#include <math.h>

#define M0 228000
#define M1 57000
#define BATCH 8
#define KPRE 2000
#define KTOT 4000
#define KPOST 1000
#define IMGW 1216.0f
#define IMGH 800.0f
#define LVLOFF 1217.0f
#define SCALE_CLAMP_F 4.135166556742356f
#define F2U_NEGINF 0x007FFFFFu

__device__ __forceinline__ unsigned f2u(float f) {
  unsigned u = __float_as_uint(f);
  return (u & 0x80000000u) ? ~u : (u | 0x80000000u);
}

// ---------------------------------------------------------------------------
// Kernel 1: exact top-KPRE per (image, level) via 4-pass byte radix select.
// grid = 16 blocks (8 images x 2 levels), 1024 threads.
// ---------------------------------------------------------------------------
__global__ __launch_bounds__(1024) void topk_kernel(
    const float* __restrict__ lg0, const float* __restrict__ lg1,
    int* __restrict__ outIdx, float* __restrict__ outScore)
{
  __shared__ unsigned hist[256];
  __shared__ unsigned shPref, shKrem, cG, cT;
  const int b   = blockIdx.x >> 1;
  const int lvl = blockIdx.x & 1;
  const int M   = lvl ? M1 : M0;
  const float* lg = lvl ? (lg1 + (size_t)b * M1) : (lg0 + (size_t)b * M0);
  const int tid = threadIdx.x;

  if (tid == 0) { shPref = 0u; shKrem = (unsigned)KPRE; cG = 0u; cT = 0u; }

  for (int p = 3; p >= 0; --p) {
    __syncthreads();                       // prev scan done / pref visible
    if (tid < 256) hist[tid] = 0u;
    __syncthreads();
    const unsigned pref = shPref;
    for (int i = tid; i < M; i += 1024) {
      unsigned u = f2u(lg[i]);
      bool ok = (p == 3) || (((u ^ pref) >> ((p + 1) * 8)) == 0u);
      if (ok) atomicAdd(&hist[(u >> (p * 8)) & 255u], 1u);
    }
    __syncthreads();
    if (tid == 0) {
      unsigned krem = shKrem, cum = 0u;
      int v;
      for (v = 255; v >= 0; --v) {
        unsigned c = hist[v];
        if (cum + c >= krem) break;
        cum += c;
      }
      if (v < 0) v = 0;                    // unreachable by invariant
      shPref = pref | ((unsigned)v << (p * 8));
      shKrem = krem - cum;
    }
  }
  __syncthreads();
  const unsigned T    = shPref;            // exact threshold key
  const unsigned krem = shKrem;            // # of ties (== T) to take
  const int base = b * KTOT + lvl * KPRE;

  for (int i = tid; i < M; i += 1024) {
    float s = lg[i];
    unsigned u = f2u(s);
    if (u > T) {
      unsigned pos = atomicAdd(&cG, 1u);   // exactly KPRE - krem of these
      outIdx[base + pos] = i;
      outScore[base + pos] = s;
    } else if (u == T) {
      unsigned pos = atomicAdd(&cT, 1u);
      if (pos < krem) {
        int slot = (KPRE - (int)krem) + (int)pos;
        outIdx[base + slot] = i;
        outScore[base + slot] = s;
      }
    }
  }
}

// ---------------------------------------------------------------------------
// Kernel 2: decode boxes for the 8*4000 selected anchors (Box2BoxTransform+clip)
// ---------------------------------------------------------------------------
__global__ void decode_kernel(
    const float* __restrict__ dl0, const float* __restrict__ dl1,
    const float* __restrict__ an0, const float* __restrict__ an1,
    const int* __restrict__ selIdx, float* __restrict__ boxes)
{
  int g = blockIdx.x * blockDim.x + threadIdx.x;
  if (g >= BATCH * KTOT) return;
  int b = g / KTOT;
  int slot = g - b * KTOT;
  int lvl = (slot >= KPRE) ? 1 : 0;
  int idx = selIdx[g];
  const float* d = lvl ? (dl1 + (((size_t)b * M1) + (size_t)idx) * 4)
                       : (dl0 + (((size_t)b * M0) + (size_t)idx) * 4);
  const float* a = lvl ? (an1 + (size_t)idx * 4) : (an0 + (size_t)idx * 4);

  float ax1 = a[0], ay1 = a[1], ax2 = a[2], ay2 = a[3];
  float w = ax2 - ax1, h = ay2 - ay1;
  float cx = ax1 + 0.5f * w, cy = ay1 + 0.5f * h;
  float dx = d[0], dy = d[1];
  float dw = fminf(d[2], SCALE_CLAMP_F);
  float dh = fminf(d[3], SCALE_CLAMP_F);
  float pcx = dx * w + cx, pcy = dy * h + cy;
  float pw = expf(dw) * w, ph = expf(dh) * h;
  float x1 = pcx - 0.5f * pw, y1 = pcy - 0.5f * ph;
  float x2 = pcx + 0.5f * pw, y2 = pcy + 0.5f * ph;
  x1 = fminf(fmaxf(x1, 0.0f), IMGW);
  x2 = fminf(fmaxf(x2, 0.0f), IMGW);
  y1 = fminf(fmaxf(y1, 0.0f), IMGH);
  y2 = fminf(fmaxf(y2, 0.0f), IMGH);
  float4 bx; bx.x = x1; bx.y = y1; bx.z = x2; bx.w = y2;
  ((float4*)boxes)[g] = bx;
}

// ---------------------------------------------------------------------------
// Kernel 3: greedy batched NMS, one workgroup per image, all state in LDS.
// gfx1250 async global->LDS staging + s_wait_asynccnt; 2 barriers/iteration;
// argmax via wave32 butterfly shuffles (all lanes get the result, no LDS
// broadcast needed).
// ---------------------------------------------------------------------------
__global__ __launch_bounds__(1024) void nms_kernel(
    const float* __restrict__ boxes, const float* __restrict__ scores,
    float* __restrict__ out)
{
  __shared__ float box4[KTOT * 4];                 // 64000 B (level-offset boxes)
  __shared__ float areaS[KTOT];                    // 16000 B
  __shared__ float sOrig[KTOT];                    // 16000 B (original scores)
  __shared__ unsigned sU[KTOT];                    // 16000 B (order-mapped keys)
  __shared__ unsigned long long red[32];

  const int b = blockIdx.x;
  const int tid = threadIdx.x;
  const int lane = tid & 31;
  const float* gb = boxes  + (size_t)b * KTOT * 4;
  const float* gs = scores + (size_t)b * KTOT;

  // --- async stage boxes (16B/slot) directly into LDS (CDNA5 path) ---
  for (int k = 0; k < 4; ++k) {
    int slot = tid + k * 1024;
    if (slot < KTOT) {
      unsigned ldsOff = (unsigned)(unsigned long long)(&box4[slot * 4]);
      unsigned long long ga = (unsigned long long)(gb + (size_t)slot * 4);
      asm volatile("global_load_async_to_lds_b128 %0, %1, off"
                   :: "v"(ldsOff), "v"(ga) : "memory");
    }
  }
  asm volatile("s_wait_asynccnt 0" ::: "memory");
  __syncthreads();

  // --- add per-level coordinate offset in place; precompute area + keys ---
  for (int k = 0; k < 4; ++k) {
    int slot = tid + k * 1024;
    if (slot < KTOT) {
      float off = (slot >= KPRE) ? LVLOFF : 0.0f;
      float x1 = box4[slot * 4 + 0] + off;
      float y1 = box4[slot * 4 + 1] + off;
      float x2 = box4[slot * 4 + 2] + off;
      float y2 = box4[slot * 4 + 3] + off;
      box4[slot * 4 + 0] = x1; box4[slot * 4 + 1] = y1;
      box4[slot * 4 + 2] = x2; box4[slot * 4 + 3] = y2;
      areaS[slot] = (x2 - x1) * (y2 - y1);
      float s = gs[slot];
      sOrig[slot] = s;
      sU[slot] = f2u(s);
    }
  }
  __syncthreads();

  for (int it = 0; it < KPOST; ++it) {
    // ---- argmax, first-occurrence tie-break: key = (u<<32)|(~slot) ----
    unsigned long long best = 0ull;   // any live key > 0 (f2u(-inf)=0x007FFFFF)
    for (int k = 0; k < 4; ++k) {
      int slot = tid + k * 1024;
      if (slot < KTOT) {
        unsigned long long key = ((unsigned long long)sU[slot] << 32)
                               | (unsigned long long)(0xFFFFFFFFu - (unsigned)slot);
        best = (key > best) ? key : best;
      }
    }
    // wave32 butterfly: every lane ends with the wave max
    for (int o = 16; o > 0; o >>= 1) {
      unsigned long long oth = __shfl_xor(best, o, 32);
      best = (oth > best) ? oth : best;
    }
    if (lane == 0) red[tid >> 5] = best;
    __syncthreads();
    // every wave redundantly reduces the 32 per-wave maxima -> all threads get j
    unsigned long long v = red[lane];
    for (int o = 16; o > 0; o >>= 1) {
      unsigned long long oth = __shfl_xor(v, o, 32);
      v = (oth > v) ? oth : v;
    }
    const unsigned j = 0xFFFFFFFFu - (unsigned)(v & 0xFFFFFFFFull);

    const float jx1 = box4[j * 4 + 0], jy1 = box4[j * 4 + 1];
    const float jx2 = box4[j * 4 + 2], jy2 = box4[j * 4 + 3];
    const float ja  = areaS[j];

    if (tid == 0) {
      float off = (j >= (unsigned)KPRE) ? LVLOFF : 0.0f;
      float* o = out + ((size_t)b * KPOST + (size_t)it) * 5;
      o[0] = jx1 - off; o[1] = jy1 - off; o[2] = jx2 - off; o[3] = jy2 - off;
      o[4] = sOrig[j];  // original (unsuppressed) score
    }

    // ---- suppress: strict iou > 0.7, plus explicit self-kill of j ----
    for (int k = 0; k < 4; ++k) {
      int slot = tid + k * 1024;
      if (slot < KTOT) {
        if ((unsigned)slot == j) {
          sU[slot] = F2U_NEGINF;
        } else {
          float ix1 = fmaxf(jx1, box4[slot * 4 + 0]);
          float iy1 = fmaxf(jy1, box4[slot * 4 + 1]);
          float ix2 = fminf(jx2, box4[slot * 4 + 2]);
          float iy2 = fminf(jy2, box4[slot * 4 + 3]);
          float inter = fmaxf(ix2 - ix1, 0.0f) * fmaxf(iy2 - iy1, 0.0f);
          float iou = inter / fmaxf(ja + areaS[slot] - inter, 1e-6f);
          if (iou > 0.7f) sU[slot] = F2U_NEGINF;
        }
      }
    }
    __syncthreads();
  }
}

// ---------------------------------------------------------------------------
extern "C" void kernel_launch(void* const* d_in, const int* in_sizes, int n_in,
                              void* d_out, int out_size, void* d_ws, size_t ws_size,
                              hipStream_t stream) {
  const float* logits0  = (const float*)d_in[0];
  const float* deltas0  = (const float*)d_in[1];
  const float* logits1  = (const float*)d_in[2];
  const float* deltas1  = (const float*)d_in[3];
  const float* anchors0 = (const float*)d_in[4];
  const float* anchors1 = (const float*)d_in[5];

  char* ws = (char*)d_ws;
  int*   wsIdx   = (int*)ws;                 // 8*4000 ints   = 128000 B
  float* wsScore = (float*)(ws + 128000);    // 8*4000 floats = 128000 B
  float* wsBoxes = (float*)(ws + 256000);    // 8*4000*4 f32  = 512000 B

  topk_kernel<<<dim3(BATCH * 2), dim3(1024), 0, stream>>>(
      logits0, logits1, wsIdx, wsScore);
  decode_kernel<<<dim3((BATCH * KTOT + 255) / 256), dim3(256), 0, stream>>>(
      deltas0, deltas1, anchors0, anchors1, wsIdx, wsBoxes);
  nms_kernel<<<dim3(BATCH), dim3(1024), 0, stream>>>(
      wsBoxes, wsScore, (float*)d_out);
}